// PruningSpanSelector_47364899340358
// MI455X (gfx1250) — compile-verified
//
#include <hip/hip_runtime.h>
#include <hip/hip_bf16.h>
#include <stdint.h>

// Problem constants (from the reference)
#define B_      8
#define T_      256
#define D_      1024
#define H_      512
#define S_TOT   32896      // T*(T+1)/2
#define KSEL    512        // pruning K
#define NGOLD   10
#define NEGV    (-1e20f)
#define CHUNK   33         // 1024 threads * 33 >= S_TOT

typedef float v2f __attribute__((ext_vector_type(2)));
typedef float v8f __attribute__((ext_vector_type(8)));

// ---------------------------------------------------------------------------
// Kernel 1: f32 projection GEMM via V_WMMA_F32_16X16X4_F32.
// Y[M,N] = X[M,Kd] * W[Kd,N] + bias ; M=B*T=2048, Kd=D=1024, N=H=512.
// One wave (32 threads) per 32(M) x 64(N) tile: 2 A-subtiles x 4 B-subtiles
// = 8 accumulators, so each B fetch group feeds 8 WMMAs (2x the FLOP per
// VMEM dword of a 16-row tile).
// A layout (ISA 7.12.2, 32-bit A 16x4): lane L -> row L%16, VGPR j holds
// K = (L/16)*2 + j.  B layout (4x16): lane L -> col L%16, VGPR j holds
// K = (L/16)*2 + j.  C/D: lane L -> col L%16, VGPR j -> row (L/16)*8 + j.
// ---------------------------------------------------------------------------
__global__ __launch_bounds__(32) void proj_gemm_kernel(
    const float* __restrict__ X, const float* __restrict__ W,
    const float* __restrict__ bias, float* __restrict__ Y)
{
    const int m0   = blockIdx.x * 32;
    const int n0   = blockIdx.y * 64;
    const int lane = threadIdx.x;
    const int mr   = lane & 15;          // row (A) / col (B,C)
    const int kk   = (lane >> 4) << 1;   // K sub-offset per half-wave

    v8f acc00 = {}, acc01 = {}, acc02 = {}, acc03 = {};
    v8f acc10 = {}, acc11 = {}, acc12 = {}, acc13 = {};
    const float* xrow0 = X + (size_t)(m0 + mr) * D_;
    const float* xrow1 = xrow0 + (size_t)16 * D_;

    for (int kb = 0; kb < D_; kb += 4) {
        v2f a0 = *reinterpret_cast<const v2f*>(xrow0 + kb + kk);
        v2f a1 = *reinterpret_cast<const v2f*>(xrow1 + kb + kk);
        const float* wp = W + (size_t)(kb + kk) * H_ + n0 + mr;
        v2f b0, b1, b2, b3;
        b0[0] = wp[0];   b0[1] = wp[H_];
        b1[0] = wp[16];  b1[1] = wp[H_ + 16];
        b2[0] = wp[32];  b2[1] = wp[H_ + 32];
        b3[0] = wp[48];  b3[1] = wp[H_ + 48];
        acc00 = __builtin_amdgcn_wmma_f32_16x16x4_f32(false, a0, false, b0, (short)0, acc00, false, false);
        acc01 = __builtin_amdgcn_wmma_f32_16x16x4_f32(false, a0, false, b1, (short)0, acc01, false, false);
        acc02 = __builtin_amdgcn_wmma_f32_16x16x4_f32(false, a0, false, b2, (short)0, acc02, false, false);
        acc03 = __builtin_amdgcn_wmma_f32_16x16x4_f32(false, a0, false, b3, (short)0, acc03, false, false);
        acc10 = __builtin_amdgcn_wmma_f32_16x16x4_f32(false, a1, false, b0, (short)0, acc10, false, false);
        acc11 = __builtin_amdgcn_wmma_f32_16x16x4_f32(false, a1, false, b1, (short)0, acc11, false, false);
        acc12 = __builtin_amdgcn_wmma_f32_16x16x4_f32(false, a1, false, b2, (short)0, acc12, false, false);
        acc13 = __builtin_amdgcn_wmma_f32_16x16x4_f32(false, a1, false, b3, (short)0, acc13, false, false);
    }

    const int mo = (lane >> 4) * 8;
    v8f accs[2][4] = {{acc00, acc01, acc02, acc03}, {acc10, acc11, acc12, acc13}};
#pragma unroll
    for (int u = 0; u < 2; ++u) {
#pragma unroll
        for (int t = 0; t < 4; ++t) {
            const int n = n0 + t * 16 + mr;
            const float bv = bias[n];
#pragma unroll
            for (int j = 0; j < 8; ++j) {
                Y[(size_t)(m0 + u * 16 + mo + j) * H_ + n] = accs[u][t][j] + bv;
            }
        }
    }
}

// ---------------------------------------------------------------------------
// Kernel 2: fused span scorer.
// score[b, tri(s,e)] = sum_h relu(hs[b,s,h] + he[b,e,h]) * w[h] + b_score
// One 256-thread block per (b, s); wave w handles ends e = s+w, s+w+8, ...
// The block first stages w_score (2KB) and the hs row (2KB) into LDS using
// GLOBAL_LOAD_ASYNC_TO_LDS_B64 (ASYNCcnt-tracked async DMA, ISA 15.18.3),
// then each lane keeps 16 h-values of each in registers via ds_load_b128.
// ---------------------------------------------------------------------------
__global__ __launch_bounds__(256) void span_score_kernel(
    const float* __restrict__ hs, const float* __restrict__ he,
    const int* __restrict__ input_mask,
    const float* __restrict__ w_score, const float* __restrict__ b_score,
    float* __restrict__ scores, unsigned char* __restrict__ smask)
{
    __shared__ float lds_w[H_];
    __shared__ float lds_hs[H_];

    const int bs   = blockIdx.x;
    const int b    = bs >> 8;      // / T_
    const int s    = bs & 255;     // % T_
    const int t    = threadIdx.x;
    const int lane = t & 31;
    const int wv   = t >> 5;       // wave 0..7

    // --- async stage: 256 threads x b64 = 2KB per array ---
    {
        const unsigned off8   = (unsigned)t * 8u;
        const unsigned ldw    = (unsigned)(uintptr_t)(&lds_w[0]) + off8;
        const unsigned ldh    = (unsigned)(uintptr_t)(&lds_hs[0]) + off8;
        const uint64_t gw     = (uint64_t)(uintptr_t)w_score;
        const uint64_t gh     = (uint64_t)(uintptr_t)(hs + ((size_t)b * T_ + s) * H_);
        asm volatile("global_load_async_to_lds_b64 %0, %1, %2"
                     :: "v"(ldw), "v"(off8), "s"(gw) : "memory");
        asm volatile("global_load_async_to_lds_b64 %0, %1, %2"
                     :: "v"(ldh), "v"(off8), "s"(gh) : "memory");
        asm volatile("s_wait_asynccnt 0" ::: "memory");
    }
    __syncthreads();

    const float4* hs4 = reinterpret_cast<const float4*>(&lds_hs[lane * 16]);
    const float4* w4  = reinterpret_cast<const float4*>(&lds_w[lane * 16]);
    float4 A0 = hs4[0], A1 = hs4[1], A2 = hs4[2], A3 = hs4[3];
    float4 W0 = w4[0],  W1 = w4[1],  W2 = w4[2],  W3 = w4[3];
    const float bsc = b_score[0];
    const int   ms  = input_mask[b * T_ + s];
    const size_t base = (size_t)b * S_TOT + (size_t)s * T_ - ((size_t)s * (s - 1)) / 2;

    for (int e = s + wv; e < T_; e += 8) {
        const float4* he4 = reinterpret_cast<const float4*>(
            he + ((size_t)b * T_ + e) * H_ + lane * 16);
        float4 E0 = he4[0], E1 = he4[1], E2 = he4[2], E3 = he4[3];
        float acc = 0.f;
        acc += fmaxf(A0.x + E0.x, 0.f) * W0.x;  acc += fmaxf(A0.y + E0.y, 0.f) * W0.y;
        acc += fmaxf(A0.z + E0.z, 0.f) * W0.z;  acc += fmaxf(A0.w + E0.w, 0.f) * W0.w;
        acc += fmaxf(A1.x + E1.x, 0.f) * W1.x;  acc += fmaxf(A1.y + E1.y, 0.f) * W1.y;
        acc += fmaxf(A1.z + E1.z, 0.f) * W1.z;  acc += fmaxf(A1.w + E1.w, 0.f) * W1.w;
        acc += fmaxf(A2.x + E2.x, 0.f) * W2.x;  acc += fmaxf(A2.y + E2.y, 0.f) * W2.y;
        acc += fmaxf(A2.z + E2.z, 0.f) * W2.z;  acc += fmaxf(A2.w + E2.w, 0.f) * W2.w;
        acc += fmaxf(A3.x + E3.x, 0.f) * W3.x;  acc += fmaxf(A3.y + E3.y, 0.f) * W3.y;
        acc += fmaxf(A3.z + E3.z, 0.f) * W3.z;  acc += fmaxf(A3.w + E3.w, 0.f) * W3.w;
        // wave32 reduction
        for (int off = 16; off > 0; off >>= 1) acc += __shfl_xor(acc, off, 32);
        if (lane == 0) {
            const int mk = ms & input_mask[b * T_ + e];
            scores[base + (e - s)] = mk ? (acc + bsc) : NEGV;
            smask [base + (e - s)] = (unsigned char)(mk ? 1 : 0);
        }
    }
}

// ---------------------------------------------------------------------------
// Kernel 3: per-batch exact top-K (radix select) with ascending index output.
// ---------------------------------------------------------------------------
__device__ __forceinline__ unsigned int key_of(float f)
{
    unsigned int u = __float_as_uint(f);
    // monotone map: bigger float -> bigger uint
    return u ^ ((((int)u) >> 31) | 0x80000000u);
}

__device__ unsigned int block_exscan(int v, volatile unsigned int* scan, int tid)
{
    scan[tid] = (unsigned int)v;
    __syncthreads();
    for (int off = 1; off < 1024; off <<= 1) {
        unsigned int t = (tid >= off) ? scan[tid - off] : 0u;
        __syncthreads();
        scan[tid] += t;
        __syncthreads();
    }
    return scan[tid] - (unsigned int)v;
}

__global__ __launch_bounds__(1024) void topk_kernel(
    const float* __restrict__ scores, int* __restrict__ top_idx)
{
    const int b = blockIdx.x;
    const float* sc = scores + (size_t)b * S_TOT;
    const int tid = threadIdx.x;

    __shared__ unsigned int hist[256];
    __shared__ unsigned int scan[1024];
    __shared__ unsigned int sh_prefix, sh_need;

    // --- 4-pass radix select of the K-th largest key ---
    unsigned int prefix = 0;
    int remaining = KSEL;
    for (int pass = 0; pass < 4; ++pass) {
        const int shift = 24 - pass * 8;
        if (tid < 256) hist[tid] = 0u;
        __syncthreads();
        for (int i = tid; i < S_TOT; i += 1024) {
            unsigned int k = key_of(sc[i]);
            if (pass == 0 || (k >> (shift + 8)) == prefix)
                atomicAdd(&hist[(k >> shift) & 255], 1u);
        }
        __syncthreads();
        if (tid == 0) {
            int rem = remaining;
            int d = 255;
            for (; d >= 0; --d) {
                int c = (int)hist[d];
                if (rem <= c) break;
                rem -= c;
            }
            sh_prefix = (prefix << 8) | (unsigned int)d;
            sh_need   = (unsigned int)rem;
        }
        __syncthreads();
        prefix    = sh_prefix;
        remaining = (int)sh_need;
        __syncthreads();
    }
    const unsigned int thrKey = prefix;   // exact key of the K-th largest
    const int need = remaining;           // # of == thrKey to take (lowest indices)

    // --- chunk statistics (contiguous chunks keep index order) ---
    const int start = tid * CHUNK;
    int eqCnt = 0, gtCnt = 0;
    for (int i = 0; i < CHUNK; ++i) {
        const int idx = start + i;
        if (idx < S_TOT) {
            const unsigned int k = key_of(sc[idx]);
            gtCnt += (k > thrKey);
            eqCnt += (k == thrKey);
        }
    }
    const unsigned int eqOff = block_exscan(eqCnt, scan, tid);
    const int eqTake = min(max(need - (int)eqOff, 0), eqCnt);
    const int selCnt = gtCnt + eqTake;
    __syncthreads();
    const unsigned int outOff = block_exscan(selCnt, scan, tid);

    // --- emit selected indices, ascending by construction ---
    int pos = (int)outOff;
    int eqSeen = 0;
    for (int i = 0; i < CHUNK; ++i) {
        const int idx = start + i;
        if (idx < S_TOT) {
            const unsigned int k = key_of(sc[idx]);
            if (k > thrKey) {
                top_idx[b * KSEL + pos++] = idx;
            } else if (k == thrKey) {
                if ((int)eqOff + eqSeen < need) top_idx[b * KSEL + pos++] = idx;
                eqSeen++;
            }
        }
    }
}

// ---------------------------------------------------------------------------
// Kernel 4: gather logits/mask, gold membership, sigmoid, per-batch BCE sum.
// ---------------------------------------------------------------------------
__global__ __launch_bounds__(KSEL) void gather_bce_kernel(
    const float* __restrict__ scores, const unsigned char* __restrict__ smask,
    const int* __restrict__ top_idx, const int* __restrict__ answer_spans,
    float* __restrict__ out_probs, float* __restrict__ partial_loss)
{
    const int b = blockIdx.x;
    const int k = threadIdx.x;
    const int idx = top_idx[b * KSEL + k];
    const float l  = scores[(size_t)b * S_TOT + idx];
    const float mk = (float)smask[(size_t)b * S_TOT + idx];

    float pred = 0.f;
#pragma unroll
    for (int g = 0; g < NGOLD; ++g) {
        const int s0 = answer_spans[(b * NGOLD + g) * 2 + 0];
        const int e0 = answer_spans[(b * NGOLD + g) * 2 + 1];
        if (s0 >= 0) {
            const int ai = (2 * s0 * T_ - s0 * s0 + s0) / 2 + (e0 - s0);
            if (ai == idx) pred = 1.f;
        }
    }

    out_probs[b * KSEL + k] = (1.f / (1.f + __expf(-l))) * mk;

    float bce = fmaxf(l, 0.f) - l * pred + log1pf(__expf(-fabsf(l)));
    bce *= mk;

    __shared__ float red[KSEL];
    red[k] = bce;
    __syncthreads();
    for (int off = KSEL / 2; off > 0; off >>= 1) {
        if (k < off) red[k] += red[k + off];
        __syncthreads();
    }
    if (k == 0) partial_loss[b] = red[0];
}

__global__ void loss_finalize_kernel(const float* __restrict__ partial,
                                     float* __restrict__ out_loss)
{
    if (threadIdx.x == 0) {
        float s = 0.f;
        for (int b = 0; b < B_; ++b) s += partial[b];
        out_loss[0] = s;
    }
}

// ---------------------------------------------------------------------------
extern "C" void kernel_launch(void* const* d_in, const int* in_sizes, int n_in,
                              void* d_out, int out_size, void* d_ws, size_t ws_size,
                              hipStream_t stream)
{
    const float* inputs       = (const float*)d_in[0];  // [B,T,D]
    const int*   input_mask   = (const int*)  d_in[1];  // [B,T]
    const int*   answer_spans = (const int*)  d_in[2];  // [B,NGOLD,2]
    const float* W_start      = (const float*)d_in[3];  // [D,H]
    const float* b_start      = (const float*)d_in[4];  // [H]
    const float* W_end        = (const float*)d_in[5];  // [D,H]
    const float* b_end        = (const float*)d_in[6];  // [H]
    const float* w_score      = (const float*)d_in[7];  // [H,1]
    const float* b_score      = (const float*)d_in[8];  // [1]

    char* ws = (char*)d_ws;
    float* hs           = (float*)(ws + 0);                 // 4 MB  [B*T, H]
    float* he           = (float*)(ws + 4194304);           // 4 MB  [B*T, H]
    float* scores       = (float*)(ws + 8388608);           // 1.05 MB [B, S_TOT]
    unsigned char* smsk = (unsigned char*)(ws + 9441280);   // 263 KB  [B, S_TOT]
    int*   top_idx      = (int*)  (ws + 9704448);           // 16 KB   [B, K]
    float* partial      = (float*)(ws + 9720832);           // 32 B    [B]

    float* out_probs = (float*)d_out;        // [B*K] = 4096
    float* out_loss  = (float*)d_out + B_ * KSEL;

    // 1) projections (WMMA f32): M=B*T=2048 -> 64 m-tiles of 32, N=512 -> 8 n-supertiles
    dim3 ggrid(64, 8, 1);
    proj_gemm_kernel<<<ggrid, 32, 0, stream>>>(inputs, W_start, b_start, hs);
    proj_gemm_kernel<<<ggrid, 32, 0, stream>>>(inputs, W_end,   b_end,   he);

    // 2) fused span scorer: one block per (b, start)
    span_score_kernel<<<B_ * T_, 256, 0, stream>>>(hs, he, input_mask,
                                                   w_score, b_score, scores, smsk);

    // 3) per-batch exact top-K with ascending index order
    topk_kernel<<<B_, 1024, 0, stream>>>(scores, top_idx);

    // 4) gather + sigmoid + per-batch BCE
    gather_bce_kernel<<<B_, KSEL, 0, stream>>>(scores, smsk, top_idx,
                                               answer_spans, out_probs, partial);

    // 5) scalar loss
    loss_finalize_kernel<<<1, 32, 0, stream>>>(partial, out_loss);
}